// ScNet_Wideband_21895743275450
// MI455X (gfx1250) — compile-verified
//
#include <hip/hip_runtime.h>
#include <hip/hip_bf16.h>
#include <cmath>

// ---------------------------------------------------------------------------
// ScNet wideband, MI455X (gfx1250, wave32).
// Shapes: BATCH=128, N=128 (features "m"/"n"), K=64, NUM_LAYER=6.
// BB/B: [b][n][m][k] fp32 (512 MB each)  -> streamed once per layer.
// ---------------------------------------------------------------------------

#define BATCH 128
#define NN    128
#define KK    64
#define NSTRIDE (NN * KK)        // stride between n-rows, in floats (8192)
#define NPAIRS (BATCH * NN)      // 16384 (b,m) pairs

typedef __attribute__((ext_vector_type(2))) float v2f;
typedef __attribute__((ext_vector_type(8))) float v8f;

// Compute 4 WMMA accumulator chains: chain c, lane L (with kl = L&15) ends up
// holding  acc_c[0] = sum_{n=0..127} x_est[b,n] * T[b, n, m, 16*c + kl]
// T is BB or B ([b][n][m][k] layout). Uses V_WMMA_F32_16X16X4_F32:
//  A (16x4 f32, replicated over the 16 M rows):
//    VGPR0 = { x[n0]   (lanes 0-15) | x[n0+2] (lanes 16-31) }
//    VGPR1 = { x[n0+1]              | x[n0+3]               }
//  B (4x16 f32), lanes = the 16 k-columns, rows in matching slots:
//    VGPR0 = { T[n0]   | T[n0+2] } ,  VGPR1 = { T[n0+1] | T[n0+3] }
// All 16 D rows are identical, so element 0 of each lane is the k-dot result.
__device__ __forceinline__ void wmma_bgemv4(
    const float* __restrict__ T, const float* __restrict__ xr,
    int b, int m, int hi, int kl,
    v8f& acc0, v8f& acc1, v8f& acc2, v8f& acc3)
{
    const size_t base0 = (((size_t)b * NN) * NN + (size_t)m) * KK; // n = 0
    #pragma unroll 2
    for (int n0 = 0; n0 < NN; n0 += 4) {
        v2f a;
        a.x = xr[n0 + 2 * hi];          // slot K0 (low) / K2 (high)
        a.y = xr[n0 + 1 + 2 * hi];      // slot K1 (low) / K3 (high)
        const float* px = T + base0 + (size_t)(n0 + 2 * hi) * NSTRIDE + kl;
        const float* py = T + base0 + (size_t)(n0 + 1 + 2 * hi) * NSTRIDE + kl;
        v2f b0, b1, b2, b3;
        b0.x = px[0];  b1.x = px[16];  b2.x = px[32];  b3.x = px[48];
        b0.y = py[0];  b1.y = py[16];  b2.y = py[32];  b3.y = py[48];
        acc0 = __builtin_amdgcn_wmma_f32_16x16x4_f32(false, a, false, b0, (short)0, acc0, false, false);
        acc1 = __builtin_amdgcn_wmma_f32_16x16x4_f32(false, a, false, b1, (short)0, acc1, false, false);
        acc2 = __builtin_amdgcn_wmma_f32_16x16x4_f32(false, a, false, b2, (short)0, acc2, false, false);
        acc3 = __builtin_amdgcn_wmma_f32_16x16x4_f32(false, a, false, b3, (short)0, acc3, false, false);
    }
}

// out_x[m][b] = x_est[b,m]*w_x[m] + sum_k (sum_n x_est[b,n]*BB[b,n,m,k] - zB[b,m,k]) * w_KL[k,m]
__global__ __launch_bounds__(256) void scnet_outx_kernel(
    const float* __restrict__ BB, const float* __restrict__ zB,
    const float* __restrict__ x_est, const float* __restrict__ w_x,
    const float* __restrict__ w_KL, float* __restrict__ out_x_T)
{
    const int lane = threadIdx.x & 31;
    const int wave = threadIdx.x >> 5;
    const int pair = blockIdx.x * 8 + wave;      // 2048 * 8 = 16384
    const int b = pair >> 7;
    const int m = pair & 127;
    const int hi = lane >> 4;
    const int kl = lane & 15;
    const float* xr = x_est + (size_t)b * NN;

    v8f a0 = {}, a1 = {}, a2 = {}, a3 = {};
    wmma_bgemv4(BB, xr, b, m, hi, kl, a0, a1, a2, a3);

    const float* zr = zB + ((size_t)b * NN + m) * KK;
    float part = (a0[0] - zr[kl])      * w_KL[(kl)      * NN + m]
               + (a1[0] - zr[16 + kl]) * w_KL[(16 + kl) * NN + m]
               + (a2[0] - zr[32 + kl]) * w_KL[(32 + kl) * NN + m]
               + (a3[0] - zr[48 + kl]) * w_KL[(48 + kl) * NN + m];
    #pragma unroll
    for (int off = 16; off > 0; off >>= 1) part += __shfl_xor(part, off, 32);
    part *= 0.5f;  // lanes 16-31 duplicate lanes 0-15
    if (lane == 0)
        out_x_T[m * BATCH + b] = xr[m] * w_x[m] + part;
}

// partials[pair] = sum_k (z[b,m,k] - sum_n x_est[b,n]*B[b,n,m,k])^2
__global__ __launch_bounds__(256) void scnet_pred_kernel(
    const float* __restrict__ Bm, const float* __restrict__ z,
    const float* __restrict__ x_est, float* __restrict__ partials)
{
    const int lane = threadIdx.x & 31;
    const int wave = threadIdx.x >> 5;
    const int pair = blockIdx.x * 8 + wave;
    const int b = pair >> 7;
    const int m = pair & 127;
    const int hi = lane >> 4;
    const int kl = lane & 15;
    const float* xr = x_est + (size_t)b * NN;

    v8f a0 = {}, a1 = {}, a2 = {}, a3 = {};
    wmma_bgemv4(Bm, xr, b, m, hi, kl, a0, a1, a2, a3);

    const float* zr = z + ((size_t)b * NN + m) * KK;
    float d0 = zr[kl]      - a0[0];
    float d1 = zr[16 + kl] - a1[0];
    float d2 = zr[32 + kl] - a2[0];
    float d3 = zr[48 + kl] - a3[0];
    float part = d0 * d0 + d1 * d1 + d2 * d2 + d3 * d3;
    #pragma unroll
    for (int off = 16; off > 0; off >>= 1) part += __shfl_xor(part, off, 32);
    part *= 0.5f;
    if (lane == 0)
        partials[pair] = part;
}

// BatchNorm1d (training: biased batch stats) + activation.
// One block per feature m; thread t = batch index.
__global__ __launch_bounds__(128) void scnet_bn_kernel(
    const float* __restrict__ out_x_T, const float* __restrict__ gamma,
    const float* __restrict__ beta, float* __restrict__ x_est_new,
    float* __restrict__ x_final, int last)
{
    const int m = blockIdx.x;
    const int t = threadIdx.x;
    const float v = out_x_T[m * BATCH + t];
    float s = v, s2 = v * v;
    #pragma unroll
    for (int off = 16; off > 0; off >>= 1) {
        s  += __shfl_xor(s,  off, 32);
        s2 += __shfl_xor(s2, off, 32);
    }
    __shared__ float ls[4], ls2[4];
    const int lane = t & 31, w = t >> 5;
    if (lane == 0) { ls[w] = s; ls2[w] = s2; }
    __syncthreads();
    s  = ls[0]  + ls[1]  + ls[2]  + ls[3];
    s2 = ls2[0] + ls2[1] + ls2[2] + ls2[3];
    const float mean = s * (1.0f / BATCH);
    const float var  = s2 * (1.0f / BATCH) - mean * mean;
    const float inv  = rsqrtf(var + 1e-5f);
    float y = gamma[m] * (v - mean) * inv + beta[m];
    if (!last) {
        y = fmaxf(y, 0.0f);
    } else {
        y = -1.0f + 2.0f * fmaxf(y + 0.5f, 0.0f) - 2.0f * fmaxf(y - 0.5f, 0.0f);
    }
    x_est_new[t * NN + m] = y;
    if (last) x_final[t * NN + m] = y;
}

// Deterministic fixed-order reduction of 16384 partials -> one loss scalar.
__global__ __launch_bounds__(256) void scnet_loss_reduce(
    const float* __restrict__ partials, float* __restrict__ out_loss, float scale)
{
    float s = 0.0f;
    for (int i = threadIdx.x; i < NPAIRS; i += 256) s += partials[i];
    #pragma unroll
    for (int off = 16; off > 0; off >>= 1) s += __shfl_xor(s, off, 32);
    __shared__ float ls[8];
    const int lane = threadIdx.x & 31, w = threadIdx.x >> 5;
    if (lane == 0) ls[w] = s;
    __syncthreads();
    if (threadIdx.x == 0) {
        float tot = 0.0f;
        #pragma unroll
        for (int i = 0; i < 8; ++i) tot += ls[i];
        out_loss[0] = scale * tot;
    }
}

extern "C" void kernel_launch(void* const* d_in, const int* in_sizes, int n_in,
                              void* d_out, int out_size, void* d_ws, size_t ws_size,
                              hipStream_t stream) {
    (void)in_sizes; (void)n_in; (void)out_size; (void)ws_size;
    const float* BB     = (const float*)d_in[0];   // [128,128,128,64]
    const float* zB     = (const float*)d_in[1];   // [128,128,64]
    /* d_in[2] = x, unused by the reference */
    const float* z      = (const float*)d_in[3];   // [128,128,64]
    const float* Bm     = (const float*)d_in[4];   // [128,128,128,64]
    const float* x_init = (const float*)d_in[5];   // [128,128]
    const float* w_x    = (const float*)d_in[6];   // [6,128]
    const float* w_KL   = (const float*)d_in[7];   // [6,64,128]
    const float* gamma  = (const float*)d_in[8];   // [6,128]
    const float* beta   = (const float*)d_in[9];   // [6,128]

    float* out = (float*)d_out;                    // [0:16384) x_est, [16384:16390) losses
    float* ws       = (float*)d_ws;
    float* out_x_T  = ws;                          // 16384 floats ([m][b])
    float* x_est    = ws + NPAIRS;                 // 16384 floats ([b][n])
    float* partials = ws + 2 * NPAIRS;             // 16384 floats

    for (int l = 0; l < 6; ++l) {
        const float* xe = (l == 0) ? x_init : x_est;
        scnet_outx_kernel<<<NPAIRS / 8, 256, 0, stream>>>(
            BB, zB, xe, w_x + l * NN, w_KL + (size_t)l * KK * NN, out_x_T);
        scnet_bn_kernel<<<NN, 128, 0, stream>>>(
            out_x_T, gamma + l * NN, beta + l * NN, x_est, out, (l == 5) ? 1 : 0);
        scnet_pred_kernel<<<NPAIRS / 8, 256, 0, stream>>>(Bm, z, x_est, partials);
        const float scale = 10.0f * (float)log((double)(l + 1)) / (float)NPAIRS;
        scnet_loss_reduce<<<1, 256, 0, stream>>>(partials, out + NPAIRS + l, scale);
    }
}